// SelectiveSSM_74543452389728
// MI455X (gfx1250) — compile-verified
//
#include <hip/hip_runtime.h>
#include <math.h>

// Problem sizes (match reference)
#define BATCH  4
#define SEQ    2048
#define DMODEL 1024
#define DSTATE 16

typedef __attribute__((ext_vector_type(2))) float v2f;
typedef __attribute__((ext_vector_type(8))) float v8f;

// 2*log2(e): folded into A / Bp so the WMMA emits the exp2 argument directly.
#define TANH_LAMBDA 2.88539008177792681472f

// Swap lane halves (lanes l <-> l^16) within wave32 via LDS lane-permute
// (no LDS memory touched).  group-of-32 encoding: xor=0x10, or=0, and=0x1f.
__device__ __forceinline__ float swapx16(float x) {
    return __builtin_bit_cast(
        float, __builtin_amdgcn_ds_swizzle(__builtin_bit_cast(int, x), 0x401F));
}

// tanh with the 2*log2(e) input scaling pre-folded into the matmul operands:
//   tanh(x) = 1 - 2 / (2^(lambda*x) + 1), here xs = lambda*x comes from WMMA.
// Saturation via exp range: exp2->inf => +1, exp2->0 => -1.  Branch-free,
// 4 ops (2 trans) on the dependent chain.
__device__ __forceinline__ float fast_tanh_pre(float xs) {
    const float t = __builtin_amdgcn_exp2f(xs);
    const float r = __builtin_amdgcn_rcpf(t + 1.0f);
    return __builtin_fmaf(-2.0f, r, 1.0f);
}

// One wave per d-channel.  Per step:
//   S_new(16x4) = tanh( A_d^T @ S + u_b * Bp_j ), y[b,t,d] = u + <S_new[:,b], C_d>
// (lambda*A^T) lives in the WMMA A-operand (constant over t), state lives in
// the B-operand, chained accumulation over four K=4 chunks.
__global__ __launch_bounds__(128) void ssm_wmma_kernel(
    const float* __restrict__ u,   // [BATCH][SEQ][DMODEL]
    const float* __restrict__ A,   // [DMODEL][DSTATE][DSTATE]
    const float* __restrict__ Bp,  // [DMODEL][DSTATE]
    const float* __restrict__ C,   // [DMODEL][DSTATE]
    float* __restrict__ y)         // [BATCH][SEQ][DMODEL]
{
    const int lane = threadIdx.x & 31;
    const int wave = threadIdx.x >> 5;
    const int d    = blockIdx.x * 4 + wave;   // one d per wave (4 waves/block)
    const int col  = lane & 15;               // N column = batch b (0..3 valid)
    const int h    = lane >> 4;               // lane half

    // ---- A-operand chunks: chunk k covers K=i in [4k,4k+4).
    // A-operand 16x4 layout: M=j = lane%16 ; VGPR c holds K_local = c + 2*h.
    // Element (M=j, K=i) = lambda * A^T[j][i] = lambda * A[d][i][j].
    v2f a[4];
#pragma unroll
    for (int k = 0; k < 4; ++k) {
#pragma unroll
        for (int c = 0; c < 2; ++c) {
            const int i = 4 * k + c + 2 * h;
            a[k][c] = TANH_LAMBDA * A[((size_t)d * DSTATE + i) * DSTATE + col];
        }
    }

    // Per-lane row constants for j = v + 8*h (C/D-layout rows).
    // bp carries the lambda pre-scale; cw is the raw output projection.
    float bp[8], cw[8];
#pragma unroll
    for (int v = 0; v < 8; ++v) {
        bp[v] = TANH_LAMBDA * Bp[(size_t)d * DSTATE + v + 8 * h];
        cw[v] = C[(size_t)d * DSTATE + v + 8 * h];
    }

    // u pointer per lane: column b = lane%16 clamped to BATCH-1 (keeps the
    // unused columns 4..15 running a bounded duplicate recurrence — no NaNs).
    const int bclamp = (col < BATCH) ? col : (BATCH - 1);
    const float* up = u + (size_t)bclamp * SEQ * DMODEL + d;
    float* yp = y + (size_t)((lane < BATCH) ? lane : 0) * SEQ * DMODEL + d;

    // State in B-operand layout: chunk k, VGPR c, lanes: i = 4k + c + 2*h.
    v2f s0 = {0.f, 0.f}, s1 = s0, s2 = s0, s3 = s0;

    // Software pipeline: u value for step t is loaded at step t-1 so the
    // s_wait_loadcnt sinks below the WMMA/tanh body instead of stalling it.
    float ut_cur = up[0];

    for (int t = 0; t < SEQ; ++t) {
        // Issue next step's load early (uniform clamped index -> SALU only).
        const int tn = (t < SEQ - 1) ? (t + 1) : t;
        const float ut_next = up[(size_t)tn * DMODEL];
        // Stream-ahead into near caches.
        __builtin_prefetch(up + (size_t)(t + 24) * DMODEL, 0, 3);

        const float ut = ut_cur;

        // C-init: (M=j, N=b) = lambda * u_b * Bp_j
        v8f acc;
#pragma unroll
        for (int v = 0; v < 8; ++v) acc[v] = ut * bp[v];

        // D = lambda * (A^T @ S + u*Bp), chained over 4 chunks of K=4.
        acc = __builtin_amdgcn_wmma_f32_16x16x4_f32(false, a[0], false, s0,
                                                    (short)0, acc, false, false);
        acc = __builtin_amdgcn_wmma_f32_16x16x4_f32(false, a[1], false, s1,
                                                    (short)0, acc, false, false);
        acc = __builtin_amdgcn_wmma_f32_16x16x4_f32(false, a[2], false, s2,
                                                    (short)0, acc, false, false);
        acc = __builtin_amdgcn_wmma_f32_16x16x4_f32(false, a[3], false, s3,
                                                    (short)0, acc, false, false);

        // Nonlinearity — branch-free, trans-pipe only (scaling pre-folded).
        float D0 = fast_tanh_pre(acc[0]), D1 = fast_tanh_pre(acc[1]);
        float D2 = fast_tanh_pre(acc[2]), D3 = fast_tanh_pre(acc[3]);
        float D4 = fast_tanh_pre(acc[4]), D5 = fast_tanh_pre(acc[5]);
        float D6 = fast_tanh_pre(acc[6]), D7 = fast_tanh_pre(acc[7]);

        // Relayout D (rows j = v + 8h) -> B-operand chunks (rows i = c + 2h'):
        // 8 of 16 rows cross lane halves -> SWAPX16 + per-half select.
        const float w0 = swapx16(D0), w1 = swapx16(D1);
        const float w2 = swapx16(D2), w3 = swapx16(D3);
        const float w4 = swapx16(D4), w5 = swapx16(D5);
        const float w6 = swapx16(D6), w7 = swapx16(D7);
        const bool lo = (h == 0);
        s0[0] = lo ? D0 : w2;  s0[1] = lo ? D1 : w3;   // i = 0,2 / 1,3
        s1[0] = lo ? D4 : w6;  s1[1] = lo ? D5 : w7;   // i = 4,6 / 5,7
        s2[0] = lo ? w0 : D2;  s2[1] = lo ? w1 : D3;   // i = 8,10 / 9,11
        s3[0] = lo ? w4 : D6;  s3[1] = lo ? w5 : D7;   // i = 12,14 / 13,15

        // y[b,t,d] = u + sum_j S_new[j,b] * C[d][j]
        float p = D0 * cw[0];
        p = fmaf(D1, cw[1], p);
        p = fmaf(D2, cw[2], p);
        p = fmaf(D3, cw[3], p);
        p = fmaf(D4, cw[4], p);
        p = fmaf(D5, cw[5], p);
        p = fmaf(D6, cw[6], p);
        p = fmaf(D7, cw[7], p);
        p += swapx16(p);  // combine j=0..7 (lo half) with j=8..15 (hi half)

        if (lane < BATCH) {
            yp[(size_t)t * DMODEL] = ut + p;
        }

        ut_cur = ut_next;  // first (and only) use of the pipelined load
    }
}

extern "C" void kernel_launch(void* const* d_in, const int* in_sizes, int n_in,
                              void* d_out, int out_size, void* d_ws,
                              size_t ws_size, hipStream_t stream) {
    (void)in_sizes; (void)n_in; (void)out_size; (void)d_ws; (void)ws_size;
    const float* u  = (const float*)d_in[0];
    const float* A  = (const float*)d_in[1];
    const float* Bp = (const float*)d_in[2];
    const float* C  = (const float*)d_in[3];
    // d_in[4] = D, unused in the reference forward pass.
    float* y = (float*)d_out;

    // 4 waves/block, one d-channel per wave: 256 blocks * 4 = DMODEL waves.
    dim3 grid(DMODEL / 4), block(128);
    hipLaunchKernelGGL(ssm_wmma_kernel, grid, block, 0, stream, u, A, Bp, C, y);
}